// GCN_44521630990730
// MI455X (gfx1250) — compile-verified
//
#include <hip/hip_runtime.h>
#include <hip/hip_bf16.h>

// GCN 2-layer forward for MI455X (gfx1250, wave32).
// GEMMs use V_WMMA_F32_16X16X4_F32 (full fp32 precision).
// Edge aggregation: wave-per-edge gather + f32 global atomics (lives in 192MB L2).

typedef __attribute__((ext_vector_type(2))) float v2f;
typedef __attribute__((ext_vector_type(8))) float v8f;

// ---------- small utility kernels ----------

__global__ void k_set1(float* __restrict__ p, int n) {
    int i = blockIdx.x * blockDim.x + threadIdx.x;
    if (i < n) p[i] = 1.0f;               // self-loop contributes 1 to every degree
}

__global__ void k_zero(float* __restrict__ p, long long n) {
    long long i = (long long)blockIdx.x * blockDim.x + threadIdx.x;
    if (i < n) p[i] = 0.0f;
}

__global__ void k_edge_deg(const int* __restrict__ dst, float* __restrict__ deg, int E) {
    int e = blockIdx.x * blockDim.x + threadIdx.x;
    if (e < E) atomicAdd(&deg[dst[e]], 1.0f);
}

__global__ void k_dinv(float* __restrict__ deg, int n) {
    int i = blockIdx.x * blockDim.x + threadIdx.x;
    if (i < n) deg[i] = rsqrtf(deg[i]);   // deg >= 1 always (self-loops)
}

// ---------- dense GEMM: H[M,NC] = A[M,K] @ W[K,NC] via fp32 WMMA ----------
// One wave computes one 16x16 tile. K chained in steps of 4.
template <int K, int NC>
__global__ void k_gemm_wmma(const float* __restrict__ A, const float* __restrict__ W,
                            float* __restrict__ H, int M) {
    const int wavesPerBlock = blockDim.x >> 5;
    int wave = blockIdx.x * wavesPerBlock + (threadIdx.x >> 5);
    int lane = threadIdx.x & 31;
    constexpr int NT = NC / 16;
    int mt = wave / NT;
    int nt = wave % NT;
    if (mt * 16 >= M) return;             // whole-wave exit; EXEC all-1s inside

    int r    = lane & 15;                 // row (A) / col (B,D) within tile
    int half = lane >> 4;                 // 0: K=k0,k0+1   1: K=k0+2,k0+3

    const float* Arow = A + (size_t)(mt * 16 + r) * K;
    const float* Wcol = W + nt * 16 + r;

    v8f acc = {};
#pragma unroll
    for (int k0 = 0; k0 < K; k0 += 4) {
        int ka = k0 + half * 2;
        v2f a;
        a[0] = Arow[ka];
        a[1] = Arow[ka + 1];
        v2f b;
        b[0] = Wcol[(size_t)ka * NC];
        b[1] = Wcol[(size_t)(ka + 1) * NC];
        acc = __builtin_amdgcn_wmma_f32_16x16x4_f32(
            /*neg_a=*/false, a, /*neg_b=*/false, b,
            /*c_mod=*/(short)0, acc, /*reuse_a=*/false, /*reuse_b=*/false);
    }

    // D layout: VGPR v -> M = v (lanes 0-15), v+8 (lanes 16-31); N = lane&15
    float* Hbase = H + (size_t)(mt * 16 + half * 8) * NC + nt * 16 + r;
#pragma unroll
    for (int v = 0; v < 8; ++v)
        Hbase[(size_t)v * NC] = acc[v];
}

// ---------- edge scatter: out[dst] += dinv[src]*dinv[dst] * h[src] ----------
// One wave per edge; lanes split channels (coalesced 128b gathers, f32 atomics).
template <int C>
__global__ void k_scatter(const float* __restrict__ h, const int* __restrict__ src,
                          const int* __restrict__ dst, const float* __restrict__ dinv,
                          float* __restrict__ out, int E) {
    constexpr int PER = C / 32;
    int wave = blockIdx.x * (blockDim.x >> 5) + (threadIdx.x >> 5);
    int lane = threadIdx.x & 31;
    if (wave >= E) return;

    int s = src[wave];
    int d = dst[wave];
    float norm = dinv[s] * dinv[d];

    const float* hs = h + (size_t)s * C + lane * PER;
    float*       od = out + (size_t)d * C + lane * PER;

    if constexpr (PER == 4) {
        float4 v = *(const float4*)hs;     // global_load_b128
        atomicAdd(od + 0, norm * v.x);
        atomicAdd(od + 1, norm * v.y);
        atomicAdd(od + 2, norm * v.z);
        atomicAdd(od + 3, norm * v.w);
    } else {
        float2 v = *(const float2*)hs;     // global_load_b64
        atomicAdd(od + 0, norm * v.x);
        atomicAdd(od + 1, norm * v.y);
    }
}

// ---------- finalize: agg += dinv[i]^2 * h[i] (self-loop) + bias, opt ReLU ----------
template <bool RELU>
__global__ void k_finalize(float* __restrict__ agg, const float* __restrict__ h,
                           const float* __restrict__ bias, const float* __restrict__ dinv,
                           int M, int C) {
    long long idx = (long long)blockIdx.x * blockDim.x + threadIdx.x;
    if (idx >= (long long)M * C) return;
    int i = (int)(idx / C);
    int c = (int)(idx % C);
    float s = dinv[i];
    float v = agg[idx] + s * s * h[idx] + bias[c];
    agg[idx] = RELU ? fmaxf(v, 0.0f) : v;
}

// ---------- host launcher ----------

extern "C" void kernel_launch(void* const* d_in, const int* in_sizes, int n_in,
                              void* d_out, int out_size, void* d_ws, size_t ws_size,
                              hipStream_t stream) {
    const float* x  = (const float*)d_in[0];   // [N,64]
    const int*   ei = (const int*)d_in[1];     // [2,E]
    const float* W1 = (const float*)d_in[2];   // [64,128]
    const float* b1 = (const float*)d_in[3];   // [128]
    const float* W2 = (const float*)d_in[4];   // [128,64]
    const float* b2 = (const float*)d_in[5];   // [64]
    float* out = (float*)d_out;                // [N,64]

    const int N = in_sizes[0] / 64;            // 100000 (multiple of 16)
    const int E = in_sizes[1] / 2;             // 3200000
    const int* src = ei;
    const int* dst = ei + E;

    // workspace: dinv[N] | h[N*128] | agg1/h1[N*128] | h2[N*64]
    float* dinv = (float*)d_ws;
    float* h    = dinv + N;
    float* agg1 = h + (size_t)N * 128;
    float* h2   = agg1 + (size_t)N * 128;

    const int T = 256;

    // --- degree / normalization ---
    k_set1<<<(N + T - 1) / T, T, 0, stream>>>(dinv, N);
    k_edge_deg<<<(E + T - 1) / T, T, 0, stream>>>(dst, dinv, E);
    k_dinv<<<(N + T - 1) / T, T, 0, stream>>>(dinv, N);

    // --- layer 1: h = x @ W1 ; agg1 = scatter(h) ; h1 = relu(agg1 + dinv^2*h + b1) ---
    long long n1 = (long long)N * 128;
    k_zero<<<(int)((n1 + T - 1) / T), T, 0, stream>>>(agg1, n1);
    {
        int tiles = (N / 16) * (128 / 16);     // 50000, divisible by 8 waves/block
        k_gemm_wmma<64, 128><<<tiles / 8, T, 0, stream>>>(x, W1, h, N);
    }
    k_scatter<128><<<(E + 7) / 8, T, 0, stream>>>(h, src, dst, dinv, agg1, E);
    k_finalize<true><<<(int)((n1 + T - 1) / T), T, 0, stream>>>(agg1, h, b1, dinv, N, 128);

    // --- layer 2: h2 = h1 @ W2 ; out = scatter(h2) + dinv^2*h2 + b2 ---
    long long n2 = (long long)N * 64;
    k_zero<<<(int)((n2 + T - 1) / T), T, 0, stream>>>(out, n2);
    {
        int tiles = (N / 16) * (64 / 16);      // 25000, divisible by 8
        k_gemm_wmma<128, 64><<<tiles / 8, T, 0, stream>>>(agg1, W2, h2, N);
    }
    k_scatter<64><<<(E + 7) / 8, T, 0, stream>>>(h2, src, dst, dinv, out, E);
    k_finalize<false><<<(int)((n2 + T - 1) / T), T, 0, stream>>>(out, h2, b2, dinv, N, 64);
}